// G2Braph_GCNConv_20469814133059
// MI455X (gfx1250) — compile-verified
//
#include <hip/hip_runtime.h>
#include <math.h>

#define N_NODES 200000
#define N_EDGES 3200000
#define D 25
#define DH 12
#define EPSV 1e-5f

typedef __attribute__((ext_vector_type(2))) float v2f;
typedef __attribute__((ext_vector_type(8))) float v8f;

// ---------------- embedding gather ----------------
__global__ void k_gather(const int* __restrict__ x, const float* __restrict__ emb,
                         float* __restrict__ h) {
    int i = blockIdx.x * blockDim.x + threadIdx.x;
    if (i < N_NODES * D) {
        int row = i / D, col = i % D;
        h[i] = emb[x[row] * D + col];
    }
}

// ---------------- degree (with self loop) ----------------
__global__ void k_deg_init(float* __restrict__ deg) {
    int i = blockIdx.x * blockDim.x + threadIdx.x;
    if (i < N_NODES) deg[i] = 1.0f;   // self loop contributes 1
}

__global__ void k_deg_accum(const int* __restrict__ dst, float* __restrict__ deg) {
    int e = blockIdx.x * blockDim.x + threadIdx.x;
    if (e < N_EDGES) atomicAdd(&deg[dst[e]], 1.0f);
}

__global__ void k_dinv(const float* __restrict__ deg, float* __restrict__ dinv) {
    int i = blockIdx.x * blockDim.x + threadIdx.x;
    if (i < N_NODES) dinv[i] = rsqrtf(fmaxf(deg[i], 1.0f));
}

// ---------------- N x 25 @ 25 x 25 GEMM via V_WMMA_F32_16X16X4_F32 ----------------
// One wave per 16-row tile. K=25 padded to 28 (7 steps of 4); two 16-col tiles
// cover the padded 32 output columns (cols >= 25 discarded on store).
// W is staged zero-padded (28x32) in LDS so B-fragment loads are unconditional
// ds_load_b32 (no exec-mask branches in the WMMA loop).
__global__ __launch_bounds__(256) void k_matmul_wmma(const float* __restrict__ h,
                                                     const float* __restrict__ W,
                                                     float* __restrict__ hW) {
    __shared__ float sW[28 * 32];
    for (int i = threadIdx.x; i < 28 * 32; i += blockDim.x) {
        int r = i >> 5, c = i & 31;
        sW[i] = (r < D && c < D) ? W[r * D + c] : 0.0f;
    }
    __syncthreads();

    const int lane = threadIdx.x & 31;
    const int wave = threadIdx.x >> 5;
    const int tile = blockIdx.x * (blockDim.x >> 5) + wave;
    const int ntiles = N_NODES / 16;   // 12500, exact
    if (tile >= ntiles) return;

    const int row0 = tile * 16;
    const int half = lane >> 4;        // 0: lanes 0-15 (K 4k,4k+1), 1: lanes 16-31 (K 4k+2,4k+3)
    const int l15  = lane & 15;
    const int arow = row0 + l15;       // A-matrix row held by this lane

    // A fragments for the 7 K-steps (shared by both column tiles).
    // For k <= 5 the guards are provably true (kb+1 <= 23 < 25); only k==6 pads.
    v2f a[7];
    #pragma unroll
    for (int k = 0; k < 7; ++k) {
        int kb = 4 * k + 2 * half;
        a[k].x = (kb     < D) ? h[arow * D + kb    ] : 0.0f;
        a[k].y = (kb + 1 < D) ? h[arow * D + kb + 1] : 0.0f;
    }

    #pragma unroll
    for (int nb = 0; nb < 32; nb += 16) {
        const int col = nb + l15;      // B/C column held by this lane
        v8f c = {0.f, 0.f, 0.f, 0.f, 0.f, 0.f, 0.f, 0.f};
        #pragma unroll
        for (int k = 0; k < 7; ++k) {
            int kb = 4 * k + 2 * half;
            v2f b;
            b.x = sW[(kb    ) * 32 + col];
            b.y = sW[(kb + 1) * 32 + col];
            c = __builtin_amdgcn_wmma_f32_16x16x4_f32(false, a[k], false, b,
                                                      (short)0, c, false, false);
        }
        if (col < D) {
            #pragma unroll
            for (int v = 0; v < 8; ++v) {
                int m = v + 8 * half;  // C layout: vgpr v holds M=v (lanes 0-15) / M=v+8 (lanes 16-31)
                hW[(row0 + m) * D + col] = c[v];
            }
        }
    }
}

// ---------------- scatter-add accumulator init: bias + self-loop message ----------------
__global__ void k_acc_init(const float* __restrict__ hW, const float* __restrict__ dinv,
                           const float* __restrict__ b, float* __restrict__ acc) {
    int i = blockIdx.x * blockDim.x + threadIdx.x;
    if (i >= N_NODES * D) return;
    int row = i / D, col = i % D;
    float w = dinv[row];
    acc[i] = b[col] + hW[i] * (w * w);
}

// ---------------- edge scatter: one wave per edge, lanes 0..24 carry features ----------------
__global__ void k_scatter(const int* __restrict__ src, const int* __restrict__ dst,
                          const float* __restrict__ dinv, const float* __restrict__ hW,
                          float* __restrict__ acc) {
    int wid  = (blockIdx.x * blockDim.x + threadIdx.x) >> 5;
    int lane = threadIdx.x & 31;
    if (wid >= N_EDGES) return;
    int s = src[wid];
    int d = dst[wid];
    float norm = dinv[s] * dinv[d];
    if (lane < D)
        atomicAdd(&acc[d * D + lane], hW[s * D + lane] * norm);
}

// ---------------- batchnorm: column sum / sumsq reduction ----------------
__global__ void k_zero_stats(float* __restrict__ csum, float* __restrict__ csq) {
    int i = threadIdx.x;
    if (i < 32) { csum[i] = 0.0f; csq[i] = 0.0f; }
}

__global__ void k_bn_reduce(const float* __restrict__ acc,
                            float* __restrict__ csum, float* __restrict__ csq) {
    __shared__ float ssum[D];
    __shared__ float ssq[D];
    if (threadIdx.x < D) { ssum[threadIdx.x] = 0.0f; ssq[threadIdx.x] = 0.0f; }
    __syncthreads();
    const int total = N_NODES * D;
    for (int i = blockIdx.x * blockDim.x + threadIdx.x; i < total;
         i += gridDim.x * blockDim.x) {
        float v = acc[i];
        int col = i % D;
        atomicAdd(&ssum[col], v);
        atomicAdd(&ssq[col], v * v);
    }
    __syncthreads();
    if (threadIdx.x < D) {
        atomicAdd(&csum[threadIdx.x], ssum[threadIdx.x]);
        atomicAdd(&csq[threadIdx.x], ssq[threadIdx.x]);
    }
}

__global__ void k_bn_apply_relu(const float* __restrict__ acc,
                                const float* __restrict__ csum, const float* __restrict__ csq,
                                const float* __restrict__ g, const float* __restrict__ be,
                                float* __restrict__ hout) {
    int i = blockIdx.x * blockDim.x + threadIdx.x;
    if (i >= N_NODES * D) return;
    int col = i % D;
    const float invN = 1.0f / (float)N_NODES;
    float mu  = csum[col] * invN;
    float var = csq[col] * invN - mu * mu;
    float sc  = rsqrtf(var + EPSV) * g[col];
    float v   = (acc[i] - mu) * sc + be[col];
    hout[i] = fmaxf(v, 0.0f);
}

// ---------------- MLP head: 25 -> 12 (relu) -> 1 (sigmoid) ----------------
__global__ void k_mlp(const float* __restrict__ h,
                      const float* __restrict__ Wm1, const float* __restrict__ bm1,
                      const float* __restrict__ Wm2, const float* __restrict__ bm2,
                      float* __restrict__ out) {
    __shared__ float sW1[D * DH];
    __shared__ float sb1[DH];
    __shared__ float sW2[DH];
    __shared__ float sb2;
    for (int i = threadIdx.x; i < D * DH; i += blockDim.x) sW1[i] = Wm1[i];
    if (threadIdx.x < DH) { sb1[threadIdx.x] = bm1[threadIdx.x]; sW2[threadIdx.x] = Wm2[threadIdx.x]; }
    if (threadIdx.x == 0) sb2 = bm2[0];
    __syncthreads();

    int n = blockIdx.x * blockDim.x + threadIdx.x;
    if (n >= N_NODES) return;

    float hr[D];
    #pragma unroll
    for (int j = 0; j < D; ++j) hr[j] = h[n * D + j];

    float z = sb2;
    #pragma unroll
    for (int o = 0; o < DH; ++o) {
        float acc = sb1[o];
        #pragma unroll
        for (int j = 0; j < D; ++j) acc = fmaf(hr[j], sW1[j * DH + o], acc);
        acc = fmaxf(acc, 0.0f);
        z = fmaf(acc, sW2[o], z);
    }
    out[n] = 1.0f / (1.0f + expf(-z));
}

// ---------------- one GCN layer ----------------
static void run_layer(const float* Wl, const float* bl, const float* gl, const float* bel,
                      float* hbuf, float* hW, float* acc,
                      const float* dinv, const int* src, const int* dst,
                      float* csum, float* csq, hipStream_t stream) {
    const int elems = N_NODES * D;
    k_matmul_wmma<<<(N_NODES / 16 + 7) / 8, 256, 0, stream>>>(hbuf, Wl, hW);
    k_acc_init<<<(elems + 255) / 256, 256, 0, stream>>>(hW, dinv, bl, acc);
    k_scatter<<<(N_EDGES + 7) / 8, 256, 0, stream>>>(src, dst, dinv, hW, acc);
    k_zero_stats<<<1, 64, 0, stream>>>(csum, csq);
    k_bn_reduce<<<1920, 256, 0, stream>>>(acc, csum, csq);
    k_bn_apply_relu<<<(elems + 255) / 256, 256, 0, stream>>>(acc, csum, csq, gl, bel, hbuf);
}

extern "C" void kernel_launch(void* const* d_in, const int* in_sizes, int n_in,
                              void* d_out, int out_size, void* d_ws, size_t ws_size,
                              hipStream_t stream) {
    const int*   x    = (const int*)d_in[0];
    const int*   ei   = (const int*)d_in[1];
    const float* emb  = (const float*)d_in[2];
    const float* W1   = (const float*)d_in[3];
    const float* b1   = (const float*)d_in[4];
    const float* g1   = (const float*)d_in[5];
    const float* be1  = (const float*)d_in[6];
    const float* W2   = (const float*)d_in[7];
    const float* b2   = (const float*)d_in[8];
    const float* g2   = (const float*)d_in[9];
    const float* be2  = (const float*)d_in[10];
    const float* Wm1  = (const float*)d_in[11];
    const float* bm1  = (const float*)d_in[12];
    const float* Wm2  = (const float*)d_in[13];
    const float* bm2  = (const float*)d_in[14];
    float* out = (float*)d_out;

    const int* src = ei;            // edge_index[0]
    const int* dst = ei + N_EDGES;  // edge_index[1]

    // workspace layout (floats): h | hW | acc | deg | dinv | csum(32) | csq(32)  (~62 MB)
    float* wsf  = (float*)d_ws;
    float* hbuf = wsf;
    float* hW   = hbuf + (size_t)N_NODES * D;
    float* acc  = hW   + (size_t)N_NODES * D;
    float* deg  = acc  + (size_t)N_NODES * D;
    float* dinv = deg  + N_NODES;
    float* csum = dinv + N_NODES;
    float* csq  = csum + 32;

    const int elems = N_NODES * D;

    k_gather<<<(elems + 255) / 256, 256, 0, stream>>>(x, emb, hbuf);
    k_deg_init<<<(N_NODES + 255) / 256, 256, 0, stream>>>(deg);
    k_deg_accum<<<(N_EDGES + 255) / 256, 256, 0, stream>>>(dst, deg);
    k_dinv<<<(N_NODES + 255) / 256, 256, 0, stream>>>(deg, dinv);

    run_layer(W1, b1, g1, be1, hbuf, hW, acc, dinv, src, dst, csum, csq, stream);
    run_layer(W2, b2, g2, be2, hbuf, hW, acc, dinv, src, dst, csum, csq, stream);

    k_mlp<<<(N_NODES + 255) / 256, 256, 0, stream>>>(hbuf, Wm1, bm1, Wm2, bm2, out);
}